// MMGATLayer_18159121727504
// MI455X (gfx1250) — compile-verified
//
#include <hip/hip_runtime.h>
#include <math.h>

#define DIM       128
#define HEADS     4
#define CPH       32          // channels per head
#define NEG_SLOPE 0.2f
#define LN_EPS    1e-5f

#define AS1 __attribute__((address_space(1)))
#define AS3 __attribute__((address_space(3)))

typedef __attribute__((ext_vector_type(2))) float v2f;
typedef __attribute__((ext_vector_type(8))) float v8f;
typedef __attribute__((ext_vector_type(4))) int   v4i;

__device__ __forceinline__ float lrelu(float v) {
    return v > 0.f ? v : NEG_SLOPE * v;
}

// Monotone float atomic max via sign-aware integer punning:
// non-negative floats order like signed ints, negative floats order
// inversely like unsigned ints. Both paths are monotone toward max.
__device__ __forceinline__ void atomicMaxFloat(float* addr, float v) {
    if (v >= 0.f) atomicMax((int*)addr, __float_as_int(v));
    else          atomicMin((unsigned int*)addr, __float_as_uint(v));
}

// ---------------------------------------------------------------------------
// 1) out = x + b   (accumulator initialization, residual + bias folded in)
// ---------------------------------------------------------------------------
__global__ void k_init_out(const float* __restrict__ x,
                           const float* __restrict__ b,
                           float* __restrict__ out, int total) {
    int i = blockIdx.x * blockDim.x + threadIdx.x;
    if (i >= total) return;
    out[i] = x[i] + b[i & (DIM - 1)];
}

// ---------------------------------------------------------------------------
// 2) xw = x @ W using V_WMMA_F32_16X16X4_F32 (exact f32 path).
//    Block = 256 threads = 8 waves. Block owns 16 rows; wave w owns cols
//    [16w,16w+16). K loop: 32 steps of K=4.
//    x tile staged to LDS via GLOBAL_LOAD_ASYNC_TO_LDS_B128 when available
//    (ASYNCcnt-tracked direct memory->LDS copy, no VGPR round trip).
//    A (16x4 f32): lanes 0-15 -> M=lane, K={0,1}; lanes 16-31 -> K={2,3}.
//    B (4x16 f32): lanes 0-15 -> N=lane, K={0,1}; lanes 16-31 -> K={2,3}.
//    D: VGPR j -> M = j (lanes 0-15) / j+8 (lanes 16-31), N = lane&15.
// ---------------------------------------------------------------------------
__global__ __launch_bounds__(256)
void k_gemm_wmma(const float* __restrict__ x, const float* __restrict__ W,
                 float* __restrict__ xw, int N) {
    __shared__ float xs[16 * 132];          // +4 pad: conflict-free column reads
    const int r0 = blockIdx.x * 16;

#if defined(__gfx1250__) && __has_builtin(__builtin_amdgcn_global_load_async_to_lds_b128)
    // 2048 floats = 512 x 16B chunks; 256 threads x 2 async b128 copies.
    // Rows are 128 floats (16B-divisible) and the padded LDS row stride
    // (132 floats = 528B) is 16B-aligned, so chunks never straddle rows.
    #pragma unroll
    for (int j = 0; j < 2; ++j) {
        int fi  = threadIdx.x * 8 + j * 4;  // float index of this 16B chunk
        int row = fi >> 7, col = fi & 127;
        int gr  = min(r0 + row, N - 1);
        __builtin_amdgcn_global_load_async_to_lds_b128(
            (AS1 v4i*)(x + (size_t)gr * DIM + col),
            (AS3 v4i*)&xs[row * 132 + col], 0, 0);
    }
  #if __has_builtin(__builtin_amdgcn_s_wait_asynccnt)
    __builtin_amdgcn_s_wait_asynccnt(0);
  #else
    asm volatile("s_wait_asynccnt 0x0" ::: "memory");
  #endif
#else
    #pragma unroll
    for (int j = 0; j < 8; ++j) {           // 2048 elems / 256 threads
        int li  = threadIdx.x + 256 * j;
        int row = li >> 7, col = li & 127;
        int gr  = min(r0 + row, N - 1);
        xs[row * 132 + col] = x[gr * DIM + col];
    }
#endif
    __syncthreads();

    const int wave = threadIdx.x >> 5;
    const int lane = threadIdx.x & 31;
    const int half = lane >> 4;             // 0: lanes 0-15, 1: lanes 16-31
    const int m    = lane & 15;
    const int col0 = wave * 16;

    v8f acc = {};
    #pragma unroll 4
    for (int kk = 0; kk < 32; ++kk) {
        const int k0 = kk * 4 + half * 2;
        v2f a, b;
        a.x = xs[m * 132 + k0];
        a.y = xs[m * 132 + k0 + 1];
        b.x = W[(k0    ) * DIM + col0 + m];
        b.y = W[(k0 + 1) * DIM + col0 + m];
        acc = __builtin_amdgcn_wmma_f32_16x16x4_f32(
                  false, a, false, b, (short)0, acc, false, false);
    }

    if (r0 + 16 <= N) {                     // full tile: unguarded stores
        #pragma unroll
        for (int j = 0; j < 8; ++j)
            xw[(r0 + j + 8 * half) * DIM + col0 + m] = acc[j];
    } else {
        #pragma unroll
        for (int j = 0; j < 8; ++j) {
            int row = r0 + j + 8 * half;
            if (row < N) xw[row * DIM + col0 + m] = acc[j];
        }
    }
}

// ---------------------------------------------------------------------------
// 3) Per-node attention dots (wave per node, wave32 shuffle reduce) and
//    amax seeded with the self-loop logit (covers every softmax segment).
// ---------------------------------------------------------------------------
__global__ __launch_bounds__(256)
void k_node_att(const float* __restrict__ xw,
                const float* __restrict__ att_src,
                const float* __restrict__ att_dst, int N,
                float* __restrict__ a_src, float* __restrict__ a_dst,
                float* __restrict__ amax) {
    int n = blockIdx.x * 8 + (threadIdx.x >> 5);
    if (n >= N) return;
    int l = threadIdx.x & 31;
    float ps[HEADS], pd[HEADS];
    #pragma unroll
    for (int h = 0; h < HEADS; ++h) {
        float v = xw[n * DIM + h * CPH + l];
        ps[h] = v * att_src[h * CPH + l];
        pd[h] = v * att_dst[h * CPH + l];
    }
    #pragma unroll
    for (int off = 16; off > 0; off >>= 1) {
        #pragma unroll
        for (int h = 0; h < HEADS; ++h) {
            ps[h] += __shfl_xor(ps[h], off, 32);
            pd[h] += __shfl_xor(pd[h], off, 32);
        }
    }
    if (l == 0) {
        #pragma unroll
        for (int h = 0; h < HEADS; ++h) {
            a_src[n * HEADS + h] = ps[h];
            a_dst[n * HEADS + h] = pd[h];
            amax [n * HEADS + h] = lrelu(ps[h] + pd[h]);   // self-loop logit
        }
    }
}

// ---------------------------------------------------------------------------
// 4) segment max over edges (float4 row loads -> single b128 per node row)
// ---------------------------------------------------------------------------
__global__ void k_edge_max(const int* __restrict__ ei, int E,
                           const float* __restrict__ a_src,
                           const float* __restrict__ a_dst,
                           float* __restrict__ amax) {
    int e = blockIdx.x * blockDim.x + threadIdx.x;
    if (e >= E) return;
    int s = ei[e], d = ei[E + e];
    const float4 s4 = ((const float4*)a_src)[s];
    const float4 d4 = ((const float4*)a_dst)[d];
    atomicMaxFloat(&amax[d * HEADS + 0], lrelu(s4.x + d4.x));
    atomicMaxFloat(&amax[d * HEADS + 1], lrelu(s4.y + d4.y));
    atomicMaxFloat(&amax[d * HEADS + 2], lrelu(s4.z + d4.z));
    atomicMaxFloat(&amax[d * HEADS + 3], lrelu(s4.w + d4.w));
}

// ---------------------------------------------------------------------------
// 5) denom init with the self-loop exp term
// ---------------------------------------------------------------------------
__global__ void k_denom_init(const float* __restrict__ a_src,
                             const float* __restrict__ a_dst,
                             const float* __restrict__ amax,
                             float* __restrict__ denom, int NH) {
    int i = blockIdx.x * blockDim.x + threadIdx.x;
    if (i >= NH) return;
    float lg = lrelu(a_src[i] + a_dst[i]);
    denom[i] = __expf(lg - amax[i]);
}

// ---------------------------------------------------------------------------
// 6) segment sum of exp terms over edges (float4 row loads)
// ---------------------------------------------------------------------------
__global__ void k_edge_sum(const int* __restrict__ ei, int E,
                           const float* __restrict__ a_src,
                           const float* __restrict__ a_dst,
                           const float* __restrict__ amax,
                           float* __restrict__ denom) {
    int e = blockIdx.x * blockDim.x + threadIdx.x;
    if (e >= E) return;
    int s = ei[e], d = ei[E + e];
    const float4 s4 = ((const float4*)a_src)[s];
    const float4 d4 = ((const float4*)a_dst)[d];
    const float4 m4 = ((const float4*)amax)[d];
    atomicAdd(&denom[d * HEADS + 0], __expf(lrelu(s4.x + d4.x) - m4.x));
    atomicAdd(&denom[d * HEADS + 1], __expf(lrelu(s4.y + d4.y) - m4.y));
    atomicAdd(&denom[d * HEADS + 2], __expf(lrelu(s4.z + d4.z) - m4.z));
    atomicAdd(&denom[d * HEADS + 3], __expf(lrelu(s4.w + d4.w) - m4.w));
}

// ---------------------------------------------------------------------------
// 7) self-loop aggregation: out[n,c] += alpha_self * xw[n,c]
// ---------------------------------------------------------------------------
__global__ void k_self_agg(const float* __restrict__ xw,
                           const float* __restrict__ a_src,
                           const float* __restrict__ a_dst,
                           const float* __restrict__ amax,
                           const float* __restrict__ denom,
                           float* __restrict__ out, int total) {
    int i = blockIdx.x * blockDim.x + threadIdx.x;
    if (i >= total) return;
    int n = i >> 7, c = i & 127, h = c >> 5;
    float lg    = lrelu(a_src[n * HEADS + h] + a_dst[n * HEADS + h]);
    float alpha = __expf(lg - amax[n * HEADS + h]) / denom[n * HEADS + h];
    out[i] += alpha * xw[i];
}

// ---------------------------------------------------------------------------
// 8) edge aggregation: 128 threads per edge -> coalesced gather of xw[src,:]
//    and coalesced GLOBAL_ATOMIC_ADD_F32 scatter into out[dst,:].
//    A 32-lane wave covers exactly one head's 32 channels, so the 4 scalar
//    softmax loads are wave-uniform (single cacheline each).
// ---------------------------------------------------------------------------
__global__ __launch_bounds__(256)
void k_edge_agg(const int* __restrict__ ei, int E,
                const float* __restrict__ xw,
                const float* __restrict__ a_src,
                const float* __restrict__ a_dst,
                const float* __restrict__ amax,
                const float* __restrict__ denom,
                float* __restrict__ out) {
    int e = blockIdx.x * 2 + (threadIdx.x >> 7);
    if (e >= E) return;
    int c = threadIdx.x & 127, h = c >> 5;
    int s = ei[e], d = ei[E + e];
    float lg    = lrelu(a_src[s * HEADS + h] + a_dst[d * HEADS + h]);
    float alpha = __expf(lg - amax[d * HEADS + h]) / denom[d * HEADS + h];
    atomicAdd(&out[d * DIM + c], alpha * xw[s * DIM + c]);
}

// ---------------------------------------------------------------------------
// 9) LayerNorm + PReLU in place (wave per row, 4 channels per lane)
// ---------------------------------------------------------------------------
__global__ __launch_bounds__(256)
void k_ln_prelu(float* __restrict__ out,
                const float* __restrict__ gamma,
                const float* __restrict__ beta,
                const float* __restrict__ prelu_w, int N) {
    int n = blockIdx.x * 8 + (threadIdx.x >> 5);
    if (n >= N) return;
    int l = threadIdx.x & 31;
    float v[4];
    float s = 0.f, s2 = 0.f;
    #pragma unroll
    for (int j = 0; j < 4; ++j) {
        v[j] = out[n * DIM + j * 32 + l];
        s  += v[j];
        s2 += v[j] * v[j];
    }
    #pragma unroll
    for (int off = 16; off > 0; off >>= 1) {
        s  += __shfl_xor(s,  off, 32);
        s2 += __shfl_xor(s2, off, 32);
    }
    const float mu  = s * (1.f / DIM);
    const float var = s2 * (1.f / DIM) - mu * mu;
    const float inv = rsqrtf(var + LN_EPS);
    const float pw  = prelu_w[0];
    #pragma unroll
    for (int j = 0; j < 4; ++j) {
        int   c = j * 32 + l;
        float h = (v[j] - mu) * inv * gamma[c] + beta[c];
        out[n * DIM + c] = h > 0.f ? h : pw * h;
    }
}

// ---------------------------------------------------------------------------
extern "C" void kernel_launch(void* const* d_in, const int* in_sizes, int n_in,
                              void* d_out, int out_size, void* d_ws, size_t ws_size,
                              hipStream_t stream) {
    const float* x        = (const float*)d_in[0];
    const int*   ei       = (const int*)  d_in[1];
    const float* W        = (const float*)d_in[2];
    const float* b        = (const float*)d_in[3];
    const float* att_src  = (const float*)d_in[4];
    const float* att_dst  = (const float*)d_in[5];
    const float* ln_gamma = (const float*)d_in[6];
    const float* ln_beta  = (const float*)d_in[7];
    const float* prelu_w  = (const float*)d_in[8];
    float* out = (float*)d_out;

    const int N  = in_sizes[0] / DIM;       // 50000
    const int E  = in_sizes[1] / 2;         // 800000
    const int NH = N * HEADS;
    const int ND = N * DIM;

    // workspace carve-out (~29 MB)
    float* xw    = (float*)d_ws;
    float* a_src = xw    + (size_t)ND;
    float* a_dst = a_src + (size_t)NH;
    float* amax  = a_dst + (size_t)NH;
    float* denom = amax  + (size_t)NH;

    k_init_out  <<<(ND + 255) / 256, 256, 0, stream>>>(x, b, out, ND);
    k_gemm_wmma <<<(N + 15) / 16,    256, 0, stream>>>(x, W, xw, N);
    k_node_att  <<<(N + 7) / 8,      256, 0, stream>>>(xw, att_src, att_dst, N,
                                                       a_src, a_dst, amax);
    k_edge_max  <<<(E + 255) / 256,  256, 0, stream>>>(ei, E, a_src, a_dst, amax);
    k_denom_init<<<(NH + 255) / 256, 256, 0, stream>>>(a_src, a_dst, amax, denom, NH);
    k_edge_sum  <<<(E + 255) / 256,  256, 0, stream>>>(ei, E, a_src, a_dst, amax, denom);
    k_self_agg  <<<(ND + 255) / 256, 256, 0, stream>>>(xw, a_src, a_dst, amax,
                                                       denom, out, ND);
    k_edge_agg  <<<(E + 1) / 2,      256, 0, stream>>>(ei, E, xw, a_src, a_dst,
                                                       amax, denom, out);
    k_ln_prelu  <<<(N + 7) / 8,      256, 0, stream>>>(out, ln_gamma, ln_beta,
                                                       prelu_w, N);
}